// Phase8Model_18313740550343
// MI455X (gfx1250) — compile-verified
//
#include <hip/hip_runtime.h>

// ---------------------------------------------------------------------------
// Phase8 energy functional on MI455X (gfx1250, wave32).
//   * pointwise/stencil reductions: one memory-bound sweep (all data L2-resident)
//   * Coulomb: two forward 3-D DFTs as twiddle-matrix matmuls on f32 WMMA
//     (V_WMMA_F32_16X16X4_F32, 48-chunk K chain per 16x16 tile), then a
//     Parseval k-space weighted dot product. No inverse FFT needed.
//   * y/x passes: one block per r-tile, 12 waves cover all k-tiles; the
//     192x16 complex input tile (24 KB) is staged into LDS ONCE per block by
//     the Tensor Data Mover (tensor_load_to_lds + s_wait_tensorcnt), killing
//     the 12x B-operand read amplification through L0/L2.
//   * deterministic: fixed-order LDS tree reductions, no float atomics.
// Workspace: ~171 MB of d_ws (tables + 6 cube-sized f32 buffers, rho reused).
// ---------------------------------------------------------------------------

#define NLEN 192
#define NSQ  (NLEN * NLEN)          // 36864
#define NT   (NLEN * NLEN * NLEN)   // 7077888
#define RBLOCKS 6912                // reduction blocks (256 thr, 4 pts/thr)
#define ZDFT_BLOCKS 3456            // 27648 tiles / 8 waves per block
#define RT_BLOCKS 2304              // r-tiles per pass (192*192/16)

typedef float v2f __attribute__((ext_vector_type(2)));
typedef float v8f __attribute__((ext_vector_type(8)));
typedef unsigned int v4u __attribute__((ext_vector_type(4)));
typedef int v4i __attribute__((ext_vector_type(4)));
typedef int v8i __attribute__((ext_vector_type(8)));

__device__ __forceinline__ int pidx(int x, int y, int z) {
    return (x * NLEN + y) * NLEN + z;
}

// ---------------------------------------------------------------------------
// Twiddle tables: W[k][n] = exp(-2*pi*i*k*n/192)  (row-major, stride 192)
// ---------------------------------------------------------------------------
__global__ void k_tables(float* __restrict__ Wre, float* __restrict__ Wim) {
    int idx = blockIdx.x * blockDim.x + threadIdx.x;   // 144*256 == 36864
    if (idx >= NSQ) return;
    int k = idx / NLEN, n = idx % NLEN;
    int prod = (k * n) % NLEN;                         // exact phase wrap
    double ang = -2.0 * 3.14159265358979323846 * (double)prod / (double)NLEN;
    Wre[idx] = (float)cos(ang);
    Wim[idx] = (float)sin(ang);
}

// ---------------------------------------------------------------------------
// Pointwise + stencil pass: emits rhoN/rhoE cubes and 11 per-block partials.
// ---------------------------------------------------------------------------
__global__ __launch_bounds__(256)
void k_pointwise(const float* __restrict__ psiN, const float* __restrict__ psiE,
                 const float* __restrict__ B0, const float* __restrict__ B1,
                 const float* __restrict__ B2,
                 float* __restrict__ rhoN_out, float* __restrict__ rhoE_out,
                 float* __restrict__ partials) {
    const float inv2dx = 1.0f / (2.0f * 0.3f);
    float s[11];
#pragma unroll
    for (int i = 0; i < 11; ++i) s[i] = 0.0f;

    int tid = threadIdx.x;
    int gid = blockIdx.x * blockDim.x + tid;
    int stride = gridDim.x * blockDim.x;

    for (int p = gid; p < NT; p += stride) {
        int z = p % NLEN, y = (p / NLEN) % NLEN, x = p / NSQ;
        int xm = (x == 0) ? NLEN - 1 : x - 1, xp = (x == NLEN - 1) ? 0 : x + 1;
        int ym = (y == 0) ? NLEN - 1 : y - 1, yp = (y == NLEN - 1) ? 0 : y + 1;
        int zm = (z == 0) ? NLEN - 1 : z - 1, zp = (z == NLEN - 1) ? 0 : z + 1;
        int iXm = pidx(xm, y, z), iXp = pidx(xp, y, z);
        int iYm = pidx(x, ym, z), iYp = pidx(x, yp, z);
        int iZm = pidx(x, y, zm), iZp = pidx(x, y, zp);

        float pn = psiN[p], pe = psiE[p];
        float b0 = B0[p], b1 = B1[p], b2 = B2[p];
        float B2s = b0 * b0 + b1 * b1 + b2 * b2;
        float rn = pn * pn + B2s;
        float re = pe * pe;
        rhoN_out[p] = rn;
        rhoE_out[p] = re;

#define GSQ(F)                                                           \
        ({ float gx_ = (F[iXp] - F[iXm]) * inv2dx;                       \
           float gy_ = (F[iYp] - F[iYm]) * inv2dx;                       \
           float gz_ = (F[iZp] - F[iZm]) * inv2dx;                       \
           gx_ * gx_ + gy_ * gy_ + gz_ * gz_; })

        s[0] += GSQ(psiN);
        s[1] += GSQ(psiE);
        s[2] += GSQ(B0) + GSQ(B1) + GSQ(B2);
        float bm = sqrtf(1e-24f + B2s) - 0.1f;
        s[3] += bm * bm;
        s[4] += rn * rn;
        s[5] += rn * rn * rn;
        s[6] += re * re;
        s[7] += re * re * re;

#define RHO_AT(i)                                                        \
        ({ float pn_ = psiN[i], c0_ = B0[i], c1_ = B1[i], c2_ = B2[i];   \
           pn_ * pn_ + c0_ * c0_ + c1_ * c1_ + c2_ * c2_; })
        float gx = (RHO_AT(iXp) - RHO_AT(iXm)) * inv2dx;
        float gy = (RHO_AT(iYp) - RHO_AT(iYm)) * inv2dx;
        float gz = (RHO_AT(iZp) - RHO_AT(iZm)) * inv2dx;
        s[8] += gx * gx + gy * gy + gz * gz;
        s[9] += rn;
        s[10] += re;
#undef GSQ
#undef RHO_AT
    }

    __shared__ float red[256];
#pragma unroll
    for (int i = 0; i < 11; ++i) {
        red[tid] = s[i];
        __syncthreads();
        for (int off = 128; off > 0; off >>= 1) {
            if (tid < off) red[tid] += red[tid + off];
            __syncthreads();
        }
        if (tid == 0) partials[blockIdx.x * 12 + i] = red[0];
        __syncthreads();
    }
}

// ---------------------------------------------------------------------------
// z-axis pass (real input, sn==1): register path, v2f B loads (contiguous K).
// One wave = one 16(k) x 16(r) tile; addr = n + (r/192)*NSQ + (r%192)*NLEN.
// ---------------------------------------------------------------------------
__global__ __launch_bounds__(256)
void k_dft_z(const float* __restrict__ Wre, const float* __restrict__ Wim,
             const float* __restrict__ inr,
             float* __restrict__ outr, float* __restrict__ outi) {
    int wid  = threadIdx.x >> 5;
    int lane = threadIdx.x & 31;
    int tile = blockIdx.x * 8 + wid;   // 27648 tiles total, exact
    int kt = tile % 12;
    int rt = tile / 12;
    int q  = rt / 12;
    int t0 = (rt % 12) * 16;
    int m  = lane & 15;
    int h  = lane >> 4;
    int k0 = kt * 16;

    long roff = (long)q * NSQ + (long)(t0 + m) * NLEN;
    const float* wr_row = Wre + (k0 + m) * NLEN;
    const float* wi_row = Wim + (k0 + m) * NLEN;
    const float* in_row = inr + roff;               // 8B aligned (even offsets)

    v8f accr = {};
    v8f acci = {};
    for (int kb = 0; kb < NLEN; kb += 4) {
        int ka = kb + 2 * h;                        // even -> 8B aligned
        v2f a_re = *(const v2f*)(wr_row + ka);
        v2f a_im = *(const v2f*)(wi_row + ka);
        v2f b_re = *(const v2f*)(in_row + ka);      // sn==1: contiguous pair
        __builtin_prefetch(in_row + ka + 4, 0, 1);  // next K chunk
        accr = __builtin_amdgcn_wmma_f32_16x16x4_f32(
            false, a_re, false, b_re, (short)0, accr, false, false);
        acci = __builtin_amdgcn_wmma_f32_16x16x4_f32(
            false, a_im, false, b_re, (short)0, acci, false, false);
    }
#pragma unroll
    for (int v = 0; v < 8; ++v) {
        int M = v + 8 * h;
        long addr = (long)(k0 + M) + roff;          // sn==1
        outr[addr] = accr[v];
        outi[addr] = acci[v];
    }
}

// ---------------------------------------------------------------------------
// TDM: stage a 2-D tile (dim0 = 16 contiguous floats, dim1 = 192 rows of
// stride sn elements) from global into LDS. Descriptor is wave-uniform.
// LDS result layout: row-packed [n][16] — exactly the B-fragment layout.
// This toolchain exposes the 6-arg builtin:
//   (uint32x4 g0, int32x8 g1, int32x4 g2, int32x4 g3, int32x8 pad, i32 cpol)
// ---------------------------------------------------------------------------
__device__ __forceinline__ void tdm_load_tile(float* ldsPtr, const float* g,
                                              int sn) {
    // generic LDS address keeps the byte offset in addr[31:0] (ISA aperture)
    unsigned lds_off = (unsigned)(unsigned long long)(void*)ldsPtr;
    unsigned long long ga = (unsigned long long)(const void*)g;
#if __has_builtin(__builtin_amdgcn_tensor_load_to_lds)
    v4u g0;
    g0[0] = 1u;                                        // count=1, user D#
    g0[1] = lds_off;                                   // lds_addr [63:32]
    g0[2] = (unsigned)(ga & 0xFFFFFFFFull);            // global_addr lo
    g0[3] = (unsigned)((ga >> 32) & 0x01FFFFFFull)     // global_addr hi
          | (2u << 30);                                // type = 2 (image)
    v8i g1;
    g1[0] = (int)(2u << 16);         // wg_mask=0 (not in cluster), 4B data
    g1[1] = (int)(16u << 16);        // abar=0 | tensor_dim0 = 16 (lo16)
    g1[2] = (int)(192u << 16);       // tensor_dim0 hi | tensor_dim1 = 192
    g1[3] = (int)(16u << 16);        // tensor_dim1 hi | tile_dim0 = 16
    g1[4] = (int)192;                // tile_dim1 = 192 | tile_dim2 = 0
    g1[5] = sn;                      // tensor_dim0_stride lo32 (elements)
    g1[6] = 0;                       // td0s hi16 | td1s lo16 (unused for 2D)
    g1[7] = 0;
    v4i zz4 = {0, 0, 0, 0};          // groups 2/3 unused (<=2D tensor)
    v8i zz8 = {0, 0, 0, 0, 0, 0, 0, 0};
    __builtin_amdgcn_tensor_load_to_lds(g0, g1, zz4, zz4, zz8, 0);
#else
    // fallback: cooperative coalesced staging (384 threads, 8 elems each)
    int tid = threadIdx.x;
    for (int idx = tid; idx < NLEN * 16; idx += 384) {
        int n = idx >> 4, j = idx & 15;
        ldsPtr[idx] = g[(long)n * sn + j];
    }
#endif
}

// ---------------------------------------------------------------------------
// y/x passes (complex, s2==1): one block per r-tile, 12 waves = 12 k-tiles.
// Input tile staged once in LDS by the TDM; all waves feed WMMA from LDS.
// B-fragment reads: lanes 0-15 hit rows ka/ka+1 (16 consecutive banks),
// lanes 16-31 rows ka+2/ka+3 (disjoint 16 banks) — conflict-free.
// ---------------------------------------------------------------------------
__global__ __launch_bounds__(384)
void k_dft_lds(const float* __restrict__ Wre, const float* __restrict__ Wim,
               const float* __restrict__ inr, const float* __restrict__ ini,
               float* __restrict__ outr, float* __restrict__ outi,
               int sn, int s1) {
    __shared__ float Bsh_re[NLEN * 16];   // 12 KB
    __shared__ float Bsh_im[NLEN * 16];   // 12 KB
    int tid = threadIdx.x;
    int wid = tid >> 5;                   // 0..11 == k-tile
    int lane = tid & 31;
    int rt = blockIdx.x;                  // 2304 r-tiles
    int q  = rt / 12;
    int t0 = (rt % 12) * 16;

    const float* src_re = inr + (long)q * s1 + t0;
    const float* src_im = ini + (long)q * s1 + t0;
#if __has_builtin(__builtin_amdgcn_tensor_load_to_lds)
    if (wid == 0) {
        tdm_load_tile(Bsh_re, src_re, sn);
        tdm_load_tile(Bsh_im, src_im, sn);
        __builtin_amdgcn_s_wait_tensorcnt(0);
    }
#else
    tdm_load_tile(Bsh_re, src_re, sn);
    tdm_load_tile(Bsh_im, src_im, sn);
#endif
    __syncthreads();

    int m = lane & 15;
    int h = lane >> 4;
    int k0 = wid * 16;
    const float* wr_row = Wre + (k0 + m) * NLEN;
    const float* wi_row = Wim + (k0 + m) * NLEN;

    v8f accr = {};
    v8f acci = {};
    for (int kb = 0; kb < NLEN; kb += 4) {
        int ka = kb + 2 * h;
        v2f a_re = *(const v2f*)(wr_row + ka);
        v2f a_im = *(const v2f*)(wi_row + ka);
        v2f b_re, b_im;
        b_re.x = Bsh_re[ka * 16 + m];
        b_re.y = Bsh_re[(ka + 1) * 16 + m];
        b_im.x = Bsh_im[ka * 16 + m];
        b_im.y = Bsh_im[(ka + 1) * 16 + m];
        v2f a_imn;
        a_imn.x = -a_im.x;                            // F32 WMMA: no A/B NEG
        a_imn.y = -a_im.y;
        accr = __builtin_amdgcn_wmma_f32_16x16x4_f32(
            false, a_re, false, b_re, (short)0, accr, false, false);
        acci = __builtin_amdgcn_wmma_f32_16x16x4_f32(
            false, a_im, false, b_re, (short)0, acci, false, false);
        accr = __builtin_amdgcn_wmma_f32_16x16x4_f32(
            false, a_imn, false, b_im, (short)0, accr, false, false);
        acci = __builtin_amdgcn_wmma_f32_16x16x4_f32(
            false, a_re, false, b_im, (short)0, acci, false, false);
    }

#pragma unroll
    for (int v = 0; v < 8; ++v) {
        int M = v + 8 * h;
        long addr = (long)(k0 + M) * sn + (long)q * s1 + (t0 + m);  // s2==1
        outr[addr] = accr[v];
        outi[addr] = acci[v];
    }
}

// ---------------------------------------------------------------------------
// k-space weighted dot: S = sum_k green(k) * (Er*Nr + Ei*Ni)
// ---------------------------------------------------------------------------
__device__ __forceinline__ float kfreq(int i) {
    float f = (i < NLEN / 2) ? (float)i : (float)(i - NLEN);
    return f * (6.283185307179586f / (NLEN * 0.3f));
}

__global__ __launch_bounds__(256)
void k_coulomb(const float* __restrict__ Er, const float* __restrict__ Ei,
               const float* __restrict__ Nr, const float* __restrict__ Ni,
               float* __restrict__ partialsC) {
    int tid = threadIdx.x;
    int gid = blockIdx.x * blockDim.x + tid;
    int stride = gridDim.x * blockDim.x;
    float s = 0.0f;
    for (int p = gid; p < NT; p += stride) {
        int z = p % NLEN, y = (p / NLEN) % NLEN, x = p / NSQ;
        float kx = kfreq(x), ky = kfreq(y), kz = kfreq(z);
        float k2 = kx * kx + ky * ky + kz * kz;
        float g = (k2 > 1e-12f) ? 1.0f / k2 : 0.0f;
        s += g * (Er[p] * Nr[p] + Ei[p] * Ni[p]);
    }
    __shared__ float red[256];
    red[tid] = s;
    __syncthreads();
    for (int off = 128; off > 0; off >>= 1) {
        if (tid < off) red[tid] += red[tid + off];
        __syncthreads();
    }
    if (tid == 0) partialsC[blockIdx.x] = red[0];
}

// ---------------------------------------------------------------------------
// Final fixed-order combine (single block -> deterministic).
// ---------------------------------------------------------------------------
__global__ __launch_bounds__(256)
void k_final(const float* __restrict__ partials1,
             const float* __restrict__ partialsC,
             float* __restrict__ out) {
    int tid = threadIdx.x;
    __shared__ float red[256];
    __shared__ float S[12];
#pragma unroll
    for (int i = 0; i < 11; ++i) {
        float loc = 0.0f;
        for (int b = tid; b < RBLOCKS; b += 256) loc += partials1[b * 12 + i];
        red[tid] = loc;
        __syncthreads();
        for (int off = 128; off > 0; off >>= 1) {
            if (tid < off) red[tid] += red[tid + off];
            __syncthreads();
        }
        if (tid == 0) S[i] = red[0];
        __syncthreads();
    }
    {
        float loc = 0.0f;
        for (int b = tid; b < RBLOCKS; b += 256) loc += partialsC[b];
        red[tid] = loc;
        __syncthreads();
        for (int off = 128; off > 0; off >>= 1) {
            if (tid < off) red[tid] += red[tid + off];
            __syncthreads();
        }
        if (tid == 0) S[11] = red[0];
        __syncthreads();
    }
    if (tid == 0) {
        const double PI = 3.14159265358979323846;
        double A13 = cbrt(56.0);
        double ISO = 26.0 * 25.0 / (A13 + 1e-12);
        double ALPHA_EFF = -2.0 + 0.01 * ISO + 0.001 * 56.0;
        double BETA_EFF = 1.0 + 0.1 * A13;
        double ALPHA_E = 0.5 * ALPHA_EFF, BETA_E = 0.5 * BETA_EFF;
        double DV = 0.3 * 0.3 * 0.3;
        double total =
            0.5 * (double)S[0] * DV
          + 0.5 * (double)S[1] * DV
          + 0.1 * (double)S[2] * DV
          + 0.05 * (double)S[3] * DV
          + 0.5 * ALPHA_EFF * (double)S[4] * DV
          + (BETA_EFF / 6.0) * (double)S[5] * DV
          + 0.5 * ALPHA_E * (double)S[6] * DV
          + (BETA_E / 6.0) * (double)S[7] * DV
          + 0.01 * (double)S[8] * DV
          + (4.0 * PI / (double)NT) * DV * (double)S[11]
          + ((double)S[9] * DV - 56.0) * ((double)S[9] * DV - 56.0)
          + ((double)S[10] * DV - 26.0) * ((double)S[10] * DV - 26.0)
          + 400.0 / A13;
        out[0] = (float)total;
    }
}

// ---------------------------------------------------------------------------
extern "C" void kernel_launch(void* const* d_in, const int* in_sizes, int n_in,
                              void* d_out, int out_size, void* d_ws,
                              size_t ws_size, hipStream_t stream) {
    const float* psiN = (const float*)d_in[0];
    const float* psiE = (const float*)d_in[1];
    const float* Bfld = (const float*)d_in[2];   // [3,192,192,192]

    char* ws = (char*)d_ws;
    float* Wre   = (float*)ws;                   // 36864 f32
    float* Wim   = Wre + NSQ;                    // 36864 f32
    float* part1 = Wim + NSQ;                    // 6912*12 f32
    float* partC = part1 + RBLOCKS * 12;         // 6912 f32
    size_t off = (size_t)1 << 20;                // cube buffers @ 1 MB
    float* rhoN = (float*)(ws + off);            // NT
    float* rhoE = rhoN + NT;                     // NT
    float* Bre  = rhoE + NT;                     // NT
    float* Bim  = Bre + NT;                      // NT
    float* Cre  = Bim + NT;                      // NT
    float* Cim  = Cre + NT;                      // NT  (total ws ~171 MB)
    float* Are  = rhoN;                          // rho dead after its pass 1
    float* Aim  = rhoE;

    k_tables<<<144, 256, 0, stream>>>(Wre, Wim);
    k_pointwise<<<RBLOCKS, 256, 0, stream>>>(psiN, psiE, Bfld, Bfld + NT,
                                             Bfld + 2 * NT, rhoN, rhoE, part1);

    // rhoE spectrum -> C   (z register path, then TDM/LDS-staged y and x)
    k_dft_z<<<ZDFT_BLOCKS, 256, 0, stream>>>(Wre, Wim, rhoE, Cre, Cim);
    k_dft_lds<<<RT_BLOCKS, 384, 0, stream>>>(Wre, Wim, Cre, Cim, Bre, Bim,
                                             NLEN, NSQ);
    k_dft_lds<<<RT_BLOCKS, 384, 0, stream>>>(Wre, Wim, Bre, Bim, Cre, Cim,
                                             NSQ, NLEN);

    // rhoN spectrum -> B   (reuses rho region as intermediate A)
    k_dft_z<<<ZDFT_BLOCKS, 256, 0, stream>>>(Wre, Wim, rhoN, Bre, Bim);
    k_dft_lds<<<RT_BLOCKS, 384, 0, stream>>>(Wre, Wim, Bre, Bim, Are, Aim,
                                             NLEN, NSQ);
    k_dft_lds<<<RT_BLOCKS, 384, 0, stream>>>(Wre, Wim, Are, Aim, Bre, Bim,
                                             NSQ, NLEN);

    k_coulomb<<<RBLOCKS, 256, 0, stream>>>(Cre, Cim, Bre, Bim, partC);
    k_final<<<1, 256, 0, stream>>>(part1, partC, (float*)d_out);
}